// MAPredNet_33887291966072
// MI455X (gfx1250) — compile-verified
//
#include <hip/hip_runtime.h>
#include <math.h>

// ---------------------------------------------------------------------------
// GraphSAGE (2 layers, mean aggr, L2-normalize, ReLU) + 2-layer MLP head.
// N=100000 nodes, E=1600000 edges, D=64. fp32 throughout; GEMMs use
// V_WMMA_F32_16X16X4_F32 (exact fp32 matrix cores on gfx1250).
// ---------------------------------------------------------------------------

typedef __attribute__((ext_vector_type(2))) float v2f;
typedef __attribute__((ext_vector_type(8))) float v8f;

#define N_NODES 100000
#define N_EDGES 1600000
#define DIM 64
#define NTILES (N_NODES / 16)   // 6250 exactly (100000 = 6250*16)
#define L2EPS 1e-12f

// ---------------------------------------------------------------------------
// WMMA wrapper: D(16x16,f32) = A(16x4,f32) * B(4x16,f32) + C
// A frag (documented layout): lane<16 holds row M=lane, {K=k0, K=k0+1};
//                             lane>=16 holds row M=lane-16, {K=k0+2, K=k0+3}.
// B frag mirrors A's K interleave; B[k][n] = W[n][k] (row-major W[out][in]),
// so lane loads W[(n0 + lane%16)*64 + k0 + 2*(lane/16)] as a float pair.
// ---------------------------------------------------------------------------
__device__ __forceinline__ v8f wmma_f32(v2f a, v2f b, v8f c) {
  return __builtin_amdgcn_wmma_f32_16x16x4_f32(
      /*neg_a=*/false, a, /*neg_b=*/false, b,
      /*c_mod=*/(short)0, c, /*reuse_a=*/false, /*reuse_b=*/false);
}

__device__ __forceinline__ v2f frag_pair(const float* __restrict__ p) {
  v2f r; r.x = p[0]; r.y = p[1]; return r;
}

// ---------------------------------------------------------------------------
// Utility kernels
// ---------------------------------------------------------------------------
__global__ void zero_kernel(float* __restrict__ p, long long n) {
  long long i = (long long)blockIdx.x * blockDim.x + threadIdx.x;
  if (i < n) p[i] = 0.0f;
}

__global__ void scatter_cnt_kernel(const int* __restrict__ dst,
                                   float* __restrict__ cnt) {
  int e = blockIdx.x * blockDim.x + threadIdx.x;
  if (e < N_EDGES) atomicAdd(&cnt[dst[e]], 1.0f);
}

// E*16 threads: each handles 4 consecutive features of one edge.
// Gather x[src] row (fully coalesced, 256B rows) and scatter-add into s[dst].
__global__ void scatter_sum_kernel(const float* __restrict__ x,
                                   const int* __restrict__ src,
                                   const int* __restrict__ dst,
                                   float* __restrict__ s) {
  long long idx = (long long)blockIdx.x * blockDim.x + threadIdx.x;
  if (idx >= (long long)N_EDGES * 16) return;
  int e  = (int)(idx >> 4);
  int d4 = ((int)idx & 15) * 4;
  int sn = src[e];
  int dn = dst[e];
  const float4 v = *(const float4*)(x + (long long)sn * DIM + d4);
  float* o = s + (long long)dn * DIM + d4;
  atomicAdd(o + 0, v.x);
  atomicAdd(o + 1, v.y);
  atomicAdd(o + 2, v.z);
  atomicAdd(o + 3, v.w);
}

// ---------------------------------------------------------------------------
// Fused SAGE layer: out = L2norm_rows(x@Wl^T + bl + (s/max(cnt,1))@Wr^T + br)
// then ReLU.  One wave32 per 16-node tile; 4 N-tiles of 16 cols each held in
// v8f accumulators; 128 fp32 WMMAs per wave.
// ---------------------------------------------------------------------------
__global__ __launch_bounds__(256) void sage_kernel(
    const float* __restrict__ xin, const float* __restrict__ ssum,
    const float* __restrict__ cnt,
    const float* __restrict__ wl, const float* __restrict__ bl,
    const float* __restrict__ wr, const float* __restrict__ br,
    float* __restrict__ out) {
  const int wave = threadIdx.x >> 5;
  const int lane = threadIdx.x & 31;
  const int tile = blockIdx.x * (blockDim.x >> 5) + wave;
  if (tile >= NTILES) return;                 // wave-uniform: EXEC stays all-1s

  const int l16  = lane & 15;
  const int half = lane >> 4;
  const int rb   = tile * 16;
  const int arow = rb + l16;                  // the row this lane owns in A

  // mean-aggregation reciprocal for this lane's row
  const float rinv = 1.0f / fmaxf(cnt[arow], 1.0f);

  // Preload all A fragments for K=0..63 (self features and prop features)
  v2f ax[16], ap[16];
#pragma unroll
  for (int k = 0; k < 16; ++k) {
    const long long off = (long long)arow * DIM + 4 * k + 2 * half;
    ax[k] = frag_pair(xin + off);
    v2f sp = frag_pair(ssum + off);
    sp.x *= rinv; sp.y *= rinv;
    ap[k] = sp;
  }

  v8f acc[4];
#pragma unroll
  for (int nt = 0; nt < 4; ++nt) {
    v8f c8 = {};
    const int n0 = nt * 16;
#pragma unroll
    for (int k = 0; k < 16; ++k) {
      v2f b = frag_pair(wl + (long long)(n0 + l16) * DIM + 4 * k + 2 * half);
      c8 = wmma_f32(ax[k], b, c8);
    }
#pragma unroll
    for (int k = 0; k < 16; ++k) {
      v2f b = frag_pair(wr + (long long)(n0 + l16) * DIM + 4 * k + 2 * half);
      c8 = wmma_f32(ap[k], b, c8);
    }
    const float bias = bl[n0 + l16] + br[n0 + l16];
#pragma unroll
    for (int r = 0; r < 8; ++r) c8[r] += bias;
    acc[nt] = c8;
  }

  // Row L2 norms. C layout: VGPR r, lanes<16 -> row rb+r, lanes>=16 -> rb+8+r.
  // Butterfly over masks 1,2,4,8 reduces within each 16-lane half.
  float ssq[8];
#pragma unroll
  for (int r = 0; r < 8; ++r) {
    float v = 0.0f;
#pragma unroll
    for (int nt = 0; nt < 4; ++nt) v += acc[nt][r] * acc[nt][r];
    ssq[r] = v;
  }
#pragma unroll
  for (int m = 1; m <= 8; m <<= 1) {
#pragma unroll
    for (int r = 0; r < 8; ++r) ssq[r] += __shfl_xor(ssq[r], m, 32);
  }

  // normalize -> ReLU -> store (coalesced: 16 lanes cover 16 consecutive cols)
#pragma unroll
  for (int r = 0; r < 8; ++r) {
    const float inv = 1.0f / fmaxf(sqrtf(ssq[r]), L2EPS);
    const int row = rb + r + 8 * half;
#pragma unroll
    for (int nt = 0; nt < 4; ++nt) {
      float v = acc[nt][r] * inv;
      v = fmaxf(v, 0.0f);
      out[(long long)row * DIM + nt * 16 + l16] = v;
    }
  }
}

// ---------------------------------------------------------------------------
// Plain linear: out = x @ W^T + b  (post-MP head, no activation)
// ---------------------------------------------------------------------------
__global__ __launch_bounds__(256) void linear_kernel(
    const float* __restrict__ xin, const float* __restrict__ w,
    const float* __restrict__ b, float* __restrict__ out) {
  const int wave = threadIdx.x >> 5;
  const int lane = threadIdx.x & 31;
  const int tile = blockIdx.x * (blockDim.x >> 5) + wave;
  if (tile >= NTILES) return;

  const int l16  = lane & 15;
  const int half = lane >> 4;
  const int rb   = tile * 16;
  const int arow = rb + l16;

  v2f ax[16];
#pragma unroll
  for (int k = 0; k < 16; ++k)
    ax[k] = frag_pair(xin + (long long)arow * DIM + 4 * k + 2 * half);

#pragma unroll
  for (int nt = 0; nt < 4; ++nt) {
    v8f c8 = {};
    const int n0 = nt * 16;
#pragma unroll
    for (int k = 0; k < 16; ++k) {
      v2f bf = frag_pair(w + (long long)(n0 + l16) * DIM + 4 * k + 2 * half);
      c8 = wmma_f32(ax[k], bf, c8);
    }
    const float bias = b[n0 + l16];
#pragma unroll
    for (int r = 0; r < 8; ++r) {
      const int row = rb + r + 8 * half;
      out[(long long)row * DIM + n0 + l16] = c8[r] + bias;
    }
  }
}

// ---------------------------------------------------------------------------
// Driver
// ---------------------------------------------------------------------------
extern "C" void kernel_launch(void* const* d_in, const int* in_sizes, int n_in,
                              void* d_out, int out_size, void* d_ws, size_t ws_size,
                              hipStream_t stream) {
  (void)in_sizes; (void)n_in; (void)out_size; (void)ws_size;

  const float* x   = (const float*)d_in[0];
  const int*   ei  = (const int*)d_in[1];       // edge_index [2, E] int32
  const int*   src = ei;
  const int*   dst = ei + N_EDGES;
  const float* w1l = (const float*)d_in[2];
  const float* b1l = (const float*)d_in[3];
  const float* w1r = (const float*)d_in[4];
  const float* b1r = (const float*)d_in[5];
  const float* w2l = (const float*)d_in[6];
  const float* b2l = (const float*)d_in[7];
  const float* w2r = (const float*)d_in[8];
  const float* b2r = (const float*)d_in[9];
  const float* wp1 = (const float*)d_in[10];
  const float* bp1 = (const float*)d_in[11];
  const float* wp2 = (const float*)d_in[12];
  const float* bp2 = (const float*)d_in[13];
  float* out = (float*)d_out;

  const long long NS = (long long)N_NODES * DIM;   // 6.4M floats
  float* s   = (float*)d_ws;       // [NS]   segment sums (reused as MLP temp)
  float* cnt = s + NS;             // [N]    in-degree counts (adjacent to s)
  float* h1  = cnt + N_NODES;      // [NS]   layer-1 output
  float* h2  = h1 + NS;            // [NS]   layer-2 output
  float* tmp = s;                  // post-MP temp reuses s
  // total ws use: (3*NS + N_NODES)*4 bytes ~= 77.2 MB

  const dim3 B(256);
  const int zgrid = (int)((NS + N_NODES + 255) / 256);
  const int egrid_cnt = (N_EDGES + 255) / 256;
  const int egrid_sum = (int)(((long long)N_EDGES * 16 + 255) / 256);
  const int ggrid = (NTILES + 7) / 8;              // 8 waves per block

  // ---- layer 1 ----
  zero_kernel<<<zgrid, B, 0, stream>>>(s, NS + N_NODES);
  scatter_cnt_kernel<<<egrid_cnt, B, 0, stream>>>(dst, cnt);
  scatter_sum_kernel<<<egrid_sum, B, 0, stream>>>(x, src, dst, s);
  sage_kernel<<<ggrid, B, 0, stream>>>(x, s, cnt, w1l, b1l, w1r, b1r, h1);

  // ---- layer 2 ----
  zero_kernel<<<zgrid, B, 0, stream>>>(s, NS + N_NODES);
  scatter_cnt_kernel<<<egrid_cnt, B, 0, stream>>>(dst, cnt);
  scatter_sum_kernel<<<egrid_sum, B, 0, stream>>>(h1, src, dst, s);
  sage_kernel<<<ggrid, B, 0, stream>>>(h1, s, cnt, w2l, b2l, w2r, b2r, h2);

  // ---- post-MP head ----
  linear_kernel<<<ggrid, B, 0, stream>>>(h2, wp1, bp1, tmp);
  linear_kernel<<<ggrid, B, 0, stream>>>(tmp, wp2, bp2, out);
}